// SatGNN_17712445128998
// MI455X (gfx1250) — compile-verified
//
#include <hip/hip_runtime.h>
#include <hip/hip_bf16.h>

// ---------------- problem constants (fixed by the reference) ----------------
#define HCH 64
#define NREL 9
#define NLAY 2
#define NEDGE 300000

__device__ __constant__ int d_relS[NREL] = {0,0,0,2,3,1,2,3,3};
__device__ __constant__ int d_relD[NREL] = {1,2,3,3,3,0,0,0,2};

static const int h_NS[4]     = {100000, 20000, 50000, 80000};
static const int h_rowoff[4] = {0, 100000, 120000, 170000};
#define TOTROWS 250000

typedef float v2f __attribute__((ext_vector_type(2)));
typedef float v8f __attribute__((ext_vector_type(8)));

// ---------------- WMMA GEMM: Y[N,64] = X[N,64] @ W[64,64] (fp32) -----------
// One wave -> 16x64 output tile via 4 accumulators, K looped in steps of 4.
// A 16x4 f32 layout: lanes 0-15 rows M, VGPR0/1 = K {0,1}; lanes 16-31 K {2,3}.
// B 4x16 f32 layout mirrors A with N across lanes. C/D: VGPR i -> M=m0+i (+8 hi).
__global__ void k_gemm64(const float* __restrict__ X, const float* __restrict__ W,
                         float* __restrict__ Y, int Nrows) {
  const int wave = threadIdx.x >> 5;
  const int lane = threadIdx.x & 31;
  const int m0 = blockIdx.x * 128 + wave * 16;
  if (m0 >= Nrows) return;                 // uniform per wave (Nrows % 16 == 0)
  const int mr = m0 + (lane & 15);
  const int kh = (lane >> 4) << 1;         // 0 or 2
  const int nc = lane & 15;
  v8f acc0 = {}, acc1 = {}, acc2 = {}, acc3 = {};
  const float* xrow = X + (size_t)mr * HCH;
  __builtin_prefetch(xrow, 0, 1);
#pragma unroll
  for (int k = 0; k < HCH; k += 4) {
    const int kk = k + kh;
    v2f a; a.x = xrow[kk]; a.y = xrow[kk + 1];
    const float* w0 = W + (size_t)kk * HCH;
    v2f b0, b1, b2, b3;
    b0.x = w0[nc];      b0.y = w0[64 + nc];
    b1.x = w0[16 + nc]; b1.y = w0[80 + nc];
    b2.x = w0[32 + nc]; b2.y = w0[96 + nc];
    b3.x = w0[48 + nc]; b3.y = w0[112 + nc];
    acc0 = __builtin_amdgcn_wmma_f32_16x16x4_f32(false, a, false, b0, (short)0, acc0, false, false);
    acc1 = __builtin_amdgcn_wmma_f32_16x16x4_f32(false, a, false, b1, (short)0, acc1, false, false);
    acc2 = __builtin_amdgcn_wmma_f32_16x16x4_f32(false, a, false, b2, (short)0, acc2, false, false);
    acc3 = __builtin_amdgcn_wmma_f32_16x16x4_f32(false, a, false, b3, (short)0, acc3, false, false);
  }
  const int mbase = m0 + ((lane >> 4) << 3);
#pragma unroll
  for (int i = 0; i < 8; ++i) {
    float* yr = Y + (size_t)(mbase + i) * HCH + nc;
    yr[0]  = acc0[i];
    yr[16] = acc1[i];
    yr[32] = acc2[i];
    yr[48] = acc3[i];
  }
}

// ---------------- fold: v[i] = dot(W[i,:], a) for src and dst --------------
__global__ void k_fold(const float* __restrict__ Ws, const float* __restrict__ as_,
                       const float* __restrict__ Wd, const float* __restrict__ ad_,
                       float* __restrict__ vv) {
  int t = threadIdx.x;
  if (t < HCH) {
    float s = 0.f;
#pragma unroll
    for (int j = 0; j < HCH; ++j) s += Ws[t * HCH + j] * as_[j];
    vv[t] = s;
  } else if (t < 2 * HCH) {
    int i = t - HCH;
    float s = 0.f;
#pragma unroll
    for (int j = 0; j < HCH; ++j) s += Wd[i * HCH + j] * ad_[j];
    vv[HCH + i] = s;
  }
}

// ---------------- per-node attention logit: al[n] = dot(X[n,:], v) ---------
__global__ void k_alpha(const float* __restrict__ X, const float* __restrict__ v,
                        float* __restrict__ al, int N) {
  int n = blockIdx.x * blockDim.x + threadIdx.x;
  if (n >= N) return;
  const float* row = X + (size_t)n * HCH;
  float s = 0.f;
#pragma unroll
  for (int i = 0; i < HCH; ++i) s += row[i] * v[i];
  al[n] = s;
}

// ---------------- per-layer bias sums per destination node type ------------
__global__ void k_bsum(const float* __restrict__ bias_l, float* __restrict__ bs) {
  int t = threadIdx.x >> 6;          // type 0..3
  int h = threadIdx.x & 63;
  float s = 0.f;
  for (int r = 0; r < NREL; ++r)
    if (d_relD[r] == t) s += bias_l[r * HCH + h];
  bs[t * HCH + h] = s;
}

__global__ void k_out_init(float* __restrict__ O, const float* __restrict__ bvec, int count) {
  int i = blockIdx.x * blockDim.x + threadIdx.x;
  if (i < count) O[i] = bvec[i & 63];
}

// ---------------- segment-softmax over edges -------------------------------
__device__ __forceinline__ unsigned enc_f32(float f) {
  unsigned u = __float_as_uint(f);
  return (u >> 31) ? ~u : (u | 0x80000000u);
}
__device__ __forceinline__ float dec_f32(unsigned e) {
  return (e >> 31) ? __uint_as_float(e & 0x7fffffffu) : __uint_as_float(~e);
}

__global__ void k_minit(unsigned* __restrict__ m, float* __restrict__ z, int N) {
  int n = blockIdx.x * blockDim.x + threadIdx.x;
  if (n < N) { m[n] = 0x007FFFFFu; /* enc(-inf) */ z[n] = 0.f; }
}

__global__ void k_score(const int* __restrict__ src, const int* __restrict__ dst,
                        const float* __restrict__ als, const float* __restrict__ ald,
                        float* __restrict__ sc, unsigned* __restrict__ m, int E) {
  int e = blockIdx.x * blockDim.x + threadIdx.x;
  if (e >= E) return;
  float v = als[src[e]] + ald[dst[e]];
  v = v > 0.f ? v : 0.2f * v;                       // leaky_relu(0.2)
  sc[e] = v;
  atomicMax(&m[dst[e]], enc_f32(v));
}

__global__ void k_exp(const int* __restrict__ dst, float* __restrict__ sc,
                      const unsigned* __restrict__ m, float* __restrict__ z, int E) {
  int e = blockIdx.x * blockDim.x + threadIdx.x;
  if (e >= E) return;
  int d = dst[e];
  float p = __expf(sc[e] - dec_f32(m[d]));
  sc[e] = p;
  atomicAdd(&z[d], p);
}

// one wave per edge: 32 lanes x 2 floats = 64-wide message scatter
__global__ void k_msg(const int* __restrict__ src, const int* __restrict__ dst,
                      const float* __restrict__ p, const float* __restrict__ z,
                      const float* __restrict__ hs, float* __restrict__ O, int E) {
  int gid = blockIdx.x * blockDim.x + threadIdx.x;
  int e = gid >> 5, lane = gid & 31;
  if (e >= E) return;
  int s = src[e], d = dst[e];
  float alpha = p[e] / (z[d] + 1e-16f);
  int h = lane << 1;
  const float* hr = hs + (size_t)s * HCH + h;
  float* orow = O + (size_t)d * HCH + h;
  atomicAdd(&orow[0], hr[0] * alpha);
  atomicAdd(&orow[1], hr[1] * alpha);
}

__global__ void k_relu(float* __restrict__ X, int count) {
  int i = blockIdx.x * blockDim.x + threadIdx.x;
  if (i < count) X[i] = fmaxf(X[i], 0.f);
}

// ---------------- pooling + head -------------------------------------------
__global__ void k_zero(float* __restrict__ p, int n) {
  int i = blockIdx.x * blockDim.x + threadIdx.x;
  if (i < n) p[i] = 0.f;
}

__global__ void k_pool(const float* __restrict__ X, const int* __restrict__ batch,
                       float* __restrict__ pool, float* __restrict__ cnt,
                       int N, int coloff) {
  int gid = blockIdx.x * blockDim.x + threadIdx.x;
  int n = gid >> 5, lane = gid & 31;
  if (n >= N) return;
  int b = batch[n];
  int h = lane << 1;
  const float* xr = X + (size_t)n * HCH + h;
  atomicAdd(&pool[b * 128 + coloff + h], xr[0]);
  atomicAdd(&pool[b * 128 + coloff + h + 1], xr[1]);
  if (lane == 0) atomicAdd(&cnt[b], 1.f);
}

__global__ void k_final(const float* __restrict__ pool, const float* __restrict__ cnt,
                        const float* __restrict__ lw, const float* __restrict__ lb,
                        float* __restrict__ out, int B) {
  int t = blockIdx.x * blockDim.x + threadIdx.x;
  if (t >= B * 2) return;
  int b = t >> 1, k = t & 1;
  float cv = fmaxf(cnt[b], 1.f);
  float cc = fmaxf(cnt[B + b], 1.f);
  float s = lb[k];
#pragma unroll 8
  for (int h = 0; h < HCH; ++h)
    s += (pool[b * 128 + h] / cv) * lw[k * 128 + h];
#pragma unroll 8
  for (int h = 0; h < HCH; ++h)
    s += (pool[b * 128 + 64 + h] / cc) * lw[k * 128 + 64 + h];
  out[b * 2 + k] = s;
}

// ---------------- host orchestration ---------------------------------------
extern "C" void kernel_launch(void* const* d_in, const int* in_sizes, int n_in,
                              void* d_out, int out_size, void* d_ws, size_t ws_size,
                              hipStream_t stream) {
  const int h_relS[NREL] = {0,0,0,2,3,1,2,3,3};
  const int h_relD[NREL] = {1,2,3,3,3,0,0,0,2};

  const float* xin[4] = {(const float*)d_in[0], (const float*)d_in[1],
                         (const float*)d_in[2], (const float*)d_in[3]};
  const float* Wsrc = (const float*)d_in[4];
  const float* Wdst = (const float*)d_in[5];
  const float* asrc = (const float*)d_in[6];
  const float* adst = (const float*)d_in[7];
  const float* bias = (const float*)d_in[8];
  const float* lin_w = (const float*)d_in[9];
  const float* lin_b = (const float*)d_in[10];
  const int* srcs[NREL]; const int* dsts[NREL];
  for (int r = 0; r < NREL; ++r) {
    srcs[r] = (const int*)d_in[11 + 2 * r];
    dsts[r] = (const int*)d_in[12 + 2 * r];
  }
  const int* batchv = (const int*)d_in[29];
  const int* batchc = (const int*)d_in[30];
  const int Bg = out_size / 2;   // 64

  // ---- workspace carve-up ----
  char* w = (char*)d_ws;
  float* XA = (float*)w;                 w += (size_t)TOTROWS * HCH * 4;
  float* XB = (float*)w;                 w += (size_t)TOTROWS * HCH * 4;
  float* HS = (float*)w;                 w += (size_t)100000 * HCH * 4;
  float* ALS = (float*)w;                w += (size_t)100000 * 4;
  float* ALD = (float*)w;                w += (size_t)100000 * 4;
  unsigned* Mb = (unsigned*)w;           w += (size_t)100000 * 4;
  float* Zb = (float*)w;                 w += (size_t)100000 * 4;
  float* Pb = (float*)w;                 w += (size_t)NEDGE * 4;
  float* VV = (float*)w;                 w += (size_t)2 * HCH * 4;
  float* BS = (float*)w;                 w += (size_t)4 * HCH * 4;
  float* POOL = (float*)w;               w += (size_t)Bg * 128 * 4;
  float* CNT = (float*)w;                w += (size_t)2 * Bg * 4;

  // ---- load node features into the ping buffer ----
  for (int t = 0; t < 4; ++t)
    hipMemcpyAsync(XA + (size_t)h_rowoff[t] * HCH, xin[t],
                   (size_t)h_NS[t] * HCH * 4, hipMemcpyDeviceToDevice, stream);

  float* cur = XA;
  float* nxt = XB;
  const int TB = 256;

  for (int l = 0; l < NLAY; ++l) {
    const float* bias_l = bias + (size_t)l * NREL * HCH;
    k_bsum<<<1, 256, 0, stream>>>(bias_l, BS);
    for (int t = 0; t < 4; ++t) {
      int cnt = h_NS[t] * HCH;
      k_out_init<<<(cnt + TB - 1) / TB, TB, 0, stream>>>(
          nxt + (size_t)h_rowoff[t] * HCH, BS + t * HCH, cnt);
    }
    for (int r = 0; r < NREL; ++r) {
      const int s = h_relS[r], d = h_relD[r];
      const int Ns = h_NS[s], Nd = h_NS[d];
      const float* Wsr = Wsrc + ((size_t)l * NREL + r) * HCH * HCH;
      const float* Wdr = Wdst + ((size_t)l * NREL + r) * HCH * HCH;
      const float* asr = asrc + ((size_t)l * NREL + r) * HCH;
      const float* adr = adst + ((size_t)l * NREL + r) * HCH;
      const float* Xs = cur + (size_t)h_rowoff[s] * HCH;
      const float* Xd = cur + (size_t)h_rowoff[d] * HCH;
      float* Od = nxt + (size_t)h_rowoff[d] * HCH;

      k_fold<<<1, 128, 0, stream>>>(Wsr, asr, Wdr, adr, VV);
      k_gemm64<<<(Ns + 127) / 128, 256, 0, stream>>>(Xs, Wsr, HS, Ns);
      k_alpha<<<(Ns + TB - 1) / TB, TB, 0, stream>>>(Xs, VV, ALS, Ns);
      k_alpha<<<(Nd + TB - 1) / TB, TB, 0, stream>>>(Xd, VV + HCH, ALD, Nd);
      k_minit<<<(Nd + TB - 1) / TB, TB, 0, stream>>>(Mb, Zb, Nd);
      k_score<<<(NEDGE + TB - 1) / TB, TB, 0, stream>>>(srcs[r], dsts[r], ALS, ALD, Pb, Mb, NEDGE);
      k_exp<<<(NEDGE + TB - 1) / TB, TB, 0, stream>>>(dsts[r], Pb, Mb, Zb, NEDGE);
      {
        long long thr = (long long)NEDGE * 32;
        k_msg<<<(unsigned)((thr + TB - 1) / TB), TB, 0, stream>>>(
            srcs[r], dsts[r], Pb, Zb, HS, Od, NEDGE);
      }
    }
    int cntAll = TOTROWS * HCH;
    k_relu<<<(cntAll + TB - 1) / TB, TB, 0, stream>>>(nxt, cntAll);
    float* tmp = cur; cur = nxt; nxt = tmp;
  }

  // ---- pooling + linear head ----
  k_zero<<<(Bg * 128 + TB - 1) / TB, TB, 0, stream>>>(POOL, Bg * 128);
  k_zero<<<(2 * Bg + TB - 1) / TB, TB, 0, stream>>>(CNT, 2 * Bg);
  {
    long long thr = (long long)h_NS[0] * 32;
    k_pool<<<(unsigned)((thr + TB - 1) / TB), TB, 0, stream>>>(
        cur, batchv, POOL, CNT, h_NS[0], 0);
    thr = (long long)h_NS[3] * 32;
    k_pool<<<(unsigned)((thr + TB - 1) / TB), TB, 0, stream>>>(
        cur + (size_t)h_rowoff[3] * HCH, batchc, POOL, CNT + Bg, h_NS[3], 64);
  }
  k_final<<<1, 128, 0, stream>>>(POOL, CNT, lin_w, lin_b, (float*)d_out, Bg);
}